// RelativeBiasGenerator_52793738003230
// MI455X (gfx1250) — compile-verified
//
#include <hip/hip_runtime.h>
#include <stdint.h>

// out[i, j] = w[j - (i + offset) + MAXLEN - 1], i in [0, seq_len), j in [0, MAXLEN)
// Pure Toeplitz broadcast: 0 FLOPs, 256 MiB of streaming writes -> write-BW bound.
// CDNA5 path: async global->LDS staging of the bias vector slice + NT b128 stores.

#define MAXLEN   8192
#define TILE_I   32
#define NTHREADS 256
#define TILE_J   (NTHREADS * 4)                   // 1024 cols per block
#define SLICE_LEN (TILE_J + TILE_I - 1)           // 1055 floats needed per block
#define SLICE_PAD (((SLICE_LEN + NTHREADS - 1) / NTHREADS) * NTHREADS)  // 1280

typedef float v4f __attribute__((ext_vector_type(4)));

__global__ __launch_bounds__(NTHREADS) void relbias_toeplitz_kernel(
    const float* __restrict__ w,      // pos_bias_w, 2*MAXLEN-1 floats
    const int*   __restrict__ d_off,  // scalar offset (device)
    float*       __restrict__ out,    // seq_len x MAXLEN
    int seq_len)
{
    __shared__ float slice[SLICE_PAD];

    const int t        = threadIdx.x;
    const int col_base = blockIdx.x * TILE_J;
    const int row_base = blockIdx.y * TILE_I;
    const int offset   = d_off[0];   // uniform scalar load

    // Smallest w-index this tile touches (may be "virtually" out of range for
    // padding rows of a partial tile; clamped per-load below):
    const int slice_base = col_base - (row_base + TILE_I - 1 + offset) + (MAXLEN - 1);

    // ---- Stage the slice into LDS with CDNA5 async global->LDS loads ----
    // Per-lane DMA: copies 4 bytes from global w + goff to the LDS address in lds_addr; tracked by ASYNCcnt.
    #pragma unroll
    for (int k = 0; k < SLICE_PAD / NTHREADS; ++k) {
        const int s  = t + k * NTHREADS;           // LDS slot (padding slots get dup data)
        int gi = slice_base + ((s < SLICE_LEN) ? s : (SLICE_LEN - 1));
        gi = gi < 0 ? 0 : (gi > 2 * MAXLEN - 2 ? 2 * MAXLEN - 2 : gi);  // always in-range for used data
        const uint32_t lds_addr = (uint32_t)(uintptr_t)(&slice[s]);     // LDS byte offset (low 32 bits)
        const uint32_t goff     = (uint32_t)gi * 4u;                    // byte offset from SGPR base
        asm volatile("global_load_async_to_lds_b32 %0, %1, %2"
                     :
                     : "v"(lds_addr), "v"(goff), "s"(w)
                     : "memory");
    }
    asm volatile("s_wait_asynccnt 0x0" ::: "memory");  // this wave's async DMAs done
    __syncthreads();                                   // all waves' DMAs visible

    // ---- Emit 32 rows of fully-coalesced, 16B-aligned non-temporal stores ----
    // Row i = row_base + r, cols [col_base + 4t, +4): LDS slot = 4t + (TILE_I-1-r).
    const int    jl   = t * 4;
    float* const orow = out + (size_t)row_base * MAXLEN + col_base + jl;

    #pragma unroll 8
    for (int r = 0; r < TILE_I; ++r) {
        const int i = row_base + r;
        if (i >= seq_len) break;
        const int s = jl + (TILE_I - 1 - r);   // arbitrary dword alignment -> gather via LDS
        v4f v;
        v.x = slice[s + 0];
        v.y = slice[s + 1];
        v.z = slice[s + 2];
        v.w = slice[s + 3];
        // 256 MiB streaming output (> 192 MB L2, never re-read): bypass with NT hint.
        __builtin_nontemporal_store(v, (v4f*)(out + (size_t)i * MAXLEN + col_base + jl));
    }
    (void)orow;
}

extern "C" void kernel_launch(void* const* d_in, const int* in_sizes, int n_in,
                              void* d_out, int out_size, void* d_ws, size_t ws_size,
                              hipStream_t stream) {
    // setup_inputs order: 0=inputs (unused), 1=seq_len, 2=offset, 3=pos_bias_w
    const float* w     = (const float*)d_in[3];
    const int*   d_off = (const int*)d_in[2];
    float*       out   = (float*)d_out;

    // seq_len is device-resident, but out_size = seq_len * MAXLEN gives it host-side.
    const int seq_len = out_size / MAXLEN;

    dim3 grid(MAXLEN / TILE_J, (seq_len + TILE_I - 1) / TILE_I);
    relbias_toeplitz_kernel<<<grid, NTHREADS, 0, stream>>>(w, d_off, out, seq_len);
}